// superglue_549755814183
// MI455X (gfx1250) — compile-verified
//
#include <hip/hip_runtime.h>
#include <math.h>

// ---------------------------------------------------------------------------
// SuperGlue-style pipeline for MI455X (gfx1250, wave32).
// ~6 MB working set -> fully L2-resident (192 MB): compute-bound problem.
// GEMMs: V_WMMA_F32_16X16X4_F32 (full f32; Sinkhorn reg=1e-3 amplifies cost
// error x1000, so low-precision WMMA is off the table; GEMM FLOPs are tiny).
// Attention: 8 dst/block, wave-per-dst, K/V staged to LDS in 64-row chunks
// via global_load_async_to_lds_b128 (ASYNCcnt) -> 8x less L2 traffic.
// ---------------------------------------------------------------------------

#define LEN     512
#define NNODE   1024
#define CH      128
#define N2      513
#define MS      516          // padded row stride of M (16B aligned)
#define REGF    0.001f
#define INVREG  1000.0f
#define LOGMASS 6.2383246250395075f   // log(512)
#define TS      64           // source rows staged per chunk (64*128*4*2 = 64KB LDS)

typedef float v2f __attribute__((ext_vector_type(2)));
typedef float v8f __attribute__((ext_vector_type(8)));

// ---- CDNA5 async global->LDS copy (ASYNCcnt path), portable inline asm ------
__device__ __forceinline__ void async_copy_b128(unsigned lds_off, const void* gaddr)
{
    asm volatile("global_load_async_to_lds_b128 %0, %1, off"
                 :: "v"(lds_off), "v"(gaddr) : "memory");
}
__device__ __forceinline__ void wait_async_all()
{
    asm volatile("s_wait_asynccnt 0x0" ::: "memory");
}

// ---------------- positional encoder: x = relu(relu(p@W1+b1)@W2+b2) + d -----
__global__ void posenc_kernel(const float* __restrict__ p1, const float* __restrict__ d1,
                              const float* __restrict__ p2, const float* __restrict__ d2,
                              const float* __restrict__ fc1w, const float* __restrict__ fc1b,
                              const float* __restrict__ fc2w, const float* __restrict__ fc2b,
                              float* __restrict__ X)
{
    int node = blockIdx.x;                    // 0..1023
    const float* p;
    const float* dsc;
    if (node < LEN) { p = p1 + node * 2;         dsc = d1 + (size_t)node * 64; }
    else            { p = p2 + (node - LEN) * 2; dsc = d2 + (size_t)(node - LEN) * 64; }

    __shared__ float h[32];
    float px = p[0], py = p[1];
    int t = threadIdx.x;                      // 64 threads
    if (t < 32) {
        float v = px * fc1w[t] + py * fc1w[32 + t] + fc1b[t];
        h[t] = fmaxf(v, 0.0f);
    }
    __syncthreads();
    float s = fc2b[t];
    #pragma unroll 8
    for (int j = 0; j < 32; ++j) s += h[j] * fc2w[j * 64 + t];
    X[(size_t)node * 64 + t] = fmaxf(s, 0.0f) + dsc[t];
}

// ---------------- f32 WMMA GEMM: C[M,N] = A[M,K] @ op(B)[K,N] (+bias, relu) --
// One wave per 16x16 tile; K stepped by 4 (V_WMMA_F32_16X16X4_F32); TRANSB is
// a template parameter so the inner loop is branch-free; all addresses are
// pointer-incremented (no per-iteration 64-bit multiplies).
template<int TRANSB>
__global__ __launch_bounds__(128)
void gemm_f32_wmma(const float* __restrict__ A, int lda,
                   const float* __restrict__ B, int ldb,
                   const float* __restrict__ bias,
                   float* __restrict__ C, int ldc,
                   int tilesM, int tilesN, int Kdim, int relu)
{
    int wave = (blockIdx.x * blockDim.x + threadIdx.x) >> 5;
    int lane = threadIdx.x & 31;
    if (wave >= tilesM * tilesN) return;      // wave-uniform guard
    int tm = wave / tilesN, tn = wave % tilesN;

    int rl  = lane & 15;
    int hi  = (lane >> 4);                    // 0: K={0,1}, 1: K={2,3}
    int row = tm * 16 + rl;
    int col = tn * 16 + rl;

    // A fragment: contiguous K pair -> single b64 load per step
    const float2* ap = (const float2*)(A + (size_t)row * lda) + hi;

    v8f acc = {};
    if (TRANSB) {
        const float2* bp = (const float2*)(B + (size_t)col * ldb) + hi;
        #pragma unroll 4
        for (int k = 0; k < Kdim; k += 4) {
            float2 af = ap[k >> 1];
            float2 bf = bp[k >> 1];
            v2f a; a.x = af.x; a.y = af.y;
            v2f b; b.x = bf.x; b.y = bf.y;
            acc = __builtin_amdgcn_wmma_f32_16x16x4_f32(
                      false, a, false, b, (short)0, acc, false, false);
        }
    } else {
        const float* bp = B + (size_t)(hi * 2) * ldb + col;
        #pragma unroll 4
        for (int k = 0; k < Kdim; k += 4) {
            float2 af = ap[k >> 1];
            v2f a; a.x = af.x; a.y = af.y;
            v2f b; b.x = bp[0]; b.y = bp[ldb];
            bp += (size_t)4 * ldb;
            acc = __builtin_amdgcn_wmma_f32_16x16x4_f32(
                      false, a, false, b, (short)0, acc, false, false);
        }
    }

    float bcol  = bias ? bias[col] : 0.0f;
    int   rbase = tm * 16 + (hi ? 8 : 0);
    #pragma unroll
    for (int i = 0; i < 8; ++i) {
        float v = acc[i] + bcol;
        if (relu) v = fmaxf(v, 0.0f);
        C[(size_t)(rbase + i) * ldc + col] = v;
    }
}

// ---------------- attentional conv: out[d] = sum_s softmax_c(q[d]*k[s])*v[s]
// Block = 256 thr = 8 waves = 8 dst nodes; each wave owns one dst (lane owns
// 4 channels -> no cross-wave reduction). K/V staged to LDS in TS-row chunks
// with global_load_async_to_lds_b128, shared by all 8 dsts (8x L2 traffic cut).
// CROSS: blocks 0..63 copy graph-1 rows (residual pass-through), 64..127 do
// attention for graph-2 dsts with graph-1 sources.
template<int CROSS>
__global__ __launch_bounds__(256)
void att_conv_kernel(const float* __restrict__ Q,
                     const float* __restrict__ Km,
                     const float* __restrict__ Vm,
                     const float* __restrict__ Hin,   // residual (null = none)
                     float* __restrict__ Hout)
{
    if (CROSS && blockIdx.x < (LEN / 8)) {    // graph-1 rows: pure residual copy
        size_t base = (size_t)blockIdx.x * 8 * CH;
        ((float4*)(Hout + base))[threadIdx.x] = ((const float4*)(Hin + base))[threadIdx.x];
        return;
    }

    __shared__ float Ks[TS * CH];
    __shared__ float Vs[TS * CH];
    unsigned ksBase = (unsigned)(size_t)(&Ks[0]);   // LDS byte offset (low 32 bits)
    unsigned vsBase = (unsigned)(size_t)(&Vs[0]);

    int w    = threadIdx.x >> 5;              // wave 0..7 -> dst
    int lane = threadIdx.x & 31;
    int d    = blockIdx.x * 8 + w;            // dst node
    int s0   = CROSS ? 0 : ((d < LEN) ? 0 : LEN);

    float4 qd = ((const float4*)(Q + (size_t)d * CH))[lane];
    float a0 = 0.f, a1 = 0.f, a2 = 0.f, a3 = 0.f;

    for (int chk = 0; chk < LEN / TS; ++chk) {
        int sc = s0 + chk * TS;
        // ---- stage K/V chunk into LDS (async, ASYNCcnt) ----
        const float4* gk = (const float4*)(Km + (size_t)sc * CH);
        const float4* gv = (const float4*)(Vm + (size_t)sc * CH);
        for (int i = threadIdx.x; i < TS * CH / 4; i += 256) {
            async_copy_b128(ksBase + (unsigned)i * 16u, gk + i);
            async_copy_b128(vsBase + (unsigned)i * 16u, gv + i);
        }
        wait_async_all();
        __syncthreads();

        // ---- per-edge channel-wise softmax + weighted accumulate ----
        for (int j = 0; j < TS; ++j) {
            int s = sc + j;
            if (!CROSS && s == d) continue;   // wave-uniform skip
            float4 kk = ((const float4*)(Ks + j * CH))[lane];
            float l0 = qd.x * kk.x, l1 = qd.y * kk.y;
            float l2 = qd.z * kk.z, l3 = qd.w * kk.w;

            float m = fmaxf(fmaxf(l0, l1), fmaxf(l2, l3));
            #pragma unroll
            for (int off = 16; off; off >>= 1) m = fmaxf(m, __shfl_xor(m, off, 32));

            float e0 = __expf(l0 - m), e1 = __expf(l1 - m);
            float e2 = __expf(l2 - m), e3 = __expf(l3 - m);
            float ssum = e0 + e1 + e2 + e3;
            #pragma unroll
            for (int off = 16; off; off >>= 1) ssum += __shfl_xor(ssum, off, 32);
            float inv = __frcp_rn(ssum);

            float4 vv = ((const float4*)(Vs + j * CH))[lane];
            a0 += e0 * inv * vv.x;  a1 += e1 * inv * vv.y;
            a2 += e2 * inv * vv.z;  a3 += e3 * inv * vv.w;
        }
        __syncthreads();                      // protect LDS before next stage
    }

    float4 res = make_float4(0.f, 0.f, 0.f, 0.f);
    if (Hin) res = ((const float4*)(Hin + (size_t)d * CH))[lane];
    ((float4*)(Hout + (size_t)d * CH))[lane] =
        make_float4(a0 + res.x, a1 + res.y, a2 + res.z, a3 + res.w);
}

// ---------------- L2 row-normalize (1024 rows x 128 ch) ----------------------
__global__ void row_normalize(float* __restrict__ H)
{
    int r = blockIdx.x, c = threadIdx.x;      // 128 threads
    float v = H[(size_t)r * CH + c];
    float s = v * v;
    #pragma unroll
    for (int off = 16; off; off >>= 1) s += __shfl_xor(s, off, 32);
    __shared__ float red[4];
    if ((c & 31) == 0) red[c >> 5] = s;
    __syncthreads();
    s = red[0] + red[1] + red[2] + red[3];
    H[(size_t)r * CH + c] = v * rsqrtf(fmaxf(s, 1e-20f));
}

// ---------------- M = 1 - [costs with dustbin row/col] -----------------------
__global__ void build_M(const float* __restrict__ costs,
                        const float* __restrict__ dustbin,
                        float* __restrict__ M)
{
    int idx = blockIdx.x * blockDim.x + threadIdx.x;
    if (idx >= N2 * N2) return;
    int i = idx / N2, j = idx % N2;
    float cwd = (i < LEN && j < LEN) ? costs[(size_t)i * LEN + j] : dustbin[0];
    M[(size_t)i * MS + j] = 1.0f - cwd;
}

__global__ void zero_fg(float* __restrict__ f, float* __restrict__ g)
{
    int t = blockIdx.x * blockDim.x + threadIdx.x;
    if (t < N2) { f[t] = 0.f; g[t] = 0.f; }
}

// ---------------- Sinkhorn half-steps (log domain) ---------------------------
__global__ void sinkhorn_row(const float* __restrict__ M, const float* __restrict__ g,
                             float* __restrict__ f)
{
    int i = blockIdx.x;
    const float* Mi = M + (size_t)i * MS;
    int lane = threadIdx.x & 31, wv = threadIdx.x >> 5;
    __shared__ float redm[4], reds[4];

    float mx = -1e30f;
    for (int j = threadIdx.x; j < N2; j += blockDim.x)
        mx = fmaxf(mx, (g[j] - Mi[j]) * INVREG);
    #pragma unroll
    for (int off = 16; off; off >>= 1) mx = fmaxf(mx, __shfl_xor(mx, off, 32));
    if (lane == 0) redm[wv] = mx;
    __syncthreads();
    mx = fmaxf(fmaxf(redm[0], redm[1]), fmaxf(redm[2], redm[3]));

    float sum = 0.f;
    for (int j = threadIdx.x; j < N2; j += blockDim.x)
        sum += __expf((g[j] - Mi[j]) * INVREG - mx);
    #pragma unroll
    for (int off = 16; off; off >>= 1) sum += __shfl_xor(sum, off, 32);
    if (lane == 0) reds[wv] = sum;
    __syncthreads();
    if (threadIdx.x == 0) {
        float loga = (i == LEN) ? LOGMASS : 0.0f;
        f[i] = REGF * (loga - (mx + __logf(reds[0] + reds[1] + reds[2] + reds[3])));
    }
}

__global__ void sinkhorn_col(const float* __restrict__ M, const float* __restrict__ f,
                             float* __restrict__ g)
{
    int j = blockIdx.x;
    int lane = threadIdx.x & 31, wv = threadIdx.x >> 5;
    __shared__ float redm[4], reds[4];

    float mx = -1e30f;
    for (int i = threadIdx.x; i < N2; i += blockDim.x)
        mx = fmaxf(mx, (f[i] - M[(size_t)i * MS + j]) * INVREG);
    #pragma unroll
    for (int off = 16; off; off >>= 1) mx = fmaxf(mx, __shfl_xor(mx, off, 32));
    if (lane == 0) redm[wv] = mx;
    __syncthreads();
    mx = fmaxf(fmaxf(redm[0], redm[1]), fmaxf(redm[2], redm[3]));

    float sum = 0.f;
    for (int i = threadIdx.x; i < N2; i += blockDim.x)
        sum += __expf((f[i] - M[(size_t)i * MS + j]) * INVREG - mx);
    #pragma unroll
    for (int off = 16; off; off >>= 1) sum += __shfl_xor(sum, off, 32);
    if (lane == 0) reds[wv] = sum;
    __syncthreads();
    if (threadIdx.x == 0) {
        float logb = (j == LEN) ? LOGMASS : 0.0f;
        g[j] = REGF * (logb - (mx + __logf(reds[0] + reds[1] + reds[2] + reds[3])));
    }
}

// ---------------- loss = mean(-log(exp((f+g-M)/reg)[matches] + 1e-3)) --------
__global__ void loss_kernel(const float* __restrict__ M, const float* __restrict__ f,
                            const float* __restrict__ g,
                            const long long* __restrict__ matches,
                            float* __restrict__ out)
{
    __shared__ float red[16];
    int t = threadIdx.x;                      // 512 threads
    long long r = matches[2 * t], c = matches[2 * t + 1];
    float pm = __expf((f[r] + g[c] - M[(size_t)r * MS + c]) * INVREG);
    float l  = -__logf(pm + 0.001f);
    #pragma unroll
    for (int off = 16; off; off >>= 1) l += __shfl_xor(l, off, 32);
    if ((t & 31) == 0) red[t >> 5] = l;
    __syncthreads();
    if (t < 16) {
        float v = red[t];
        #pragma unroll
        for (int off = 8; off; off >>= 1) v += __shfl_xor(v, off, 16);
        if (t == 0) out[0] = v / 512.0f;
    }
}

// ---------------------------------------------------------------------------
extern "C" void kernel_launch(void* const* d_in, const int* in_sizes, int n_in,
                              void* d_out, int out_size, void* d_ws, size_t ws_size,
                              hipStream_t stream)
{
    const float*     p1      = (const float*)d_in[0];
    const float*     d1      = (const float*)d_in[1];
    const float*     p2      = (const float*)d_in[2];
    const float*     d2      = (const float*)d_in[3];
    const long long* matches = (const long long*)d_in[4];   // int64 per reference

    // Detect pytree flatten order (insertion vs jax sorted-key) via sizes:
    // sorted order puts 'dustbin' (size 1) at index 5.
    const float *fc1w, *fc1b, *fc2w, *fc2b, *fc3w, *fc3b, *dustbin;
    const float *W[4][3], *Bi[4][3];          // per layer: W1,W2,W3 / b1,b2,b3
    if (in_sizes[5] == 1) {                   // jax sorted: dustbin,fc*_b,fc*_w,mp
        dustbin = (const float*)d_in[5];
        fc1b = (const float*)d_in[6];  fc1w = (const float*)d_in[7];
        fc2b = (const float*)d_in[8];  fc2w = (const float*)d_in[9];
        fc3b = (const float*)d_in[10]; fc3w = (const float*)d_in[11];
        int idx = 12;
        for (int l = 0; l < 4; ++l) {         // W1,W2,W3,b1,b2,b3 (ASCII sort)
            W[l][0] = (const float*)d_in[idx + 0];
            W[l][1] = (const float*)d_in[idx + 1];
            W[l][2] = (const float*)d_in[idx + 2];
            Bi[l][0] = (const float*)d_in[idx + 3];
            Bi[l][1] = (const float*)d_in[idx + 4];
            Bi[l][2] = (const float*)d_in[idx + 5];
            idx += 6;
        }
    } else {                                  // insertion order
        fc1w = (const float*)d_in[5];  fc1b = (const float*)d_in[6];
        fc2w = (const float*)d_in[7];  fc2b = (const float*)d_in[8];
        fc3w = (const float*)d_in[9];  fc3b = (const float*)d_in[10];
        int idx = 11;
        for (int l = 0; l < 4; ++l) {         // W1,b1,W2,b2,W3,b3
            W[l][0] = (const float*)d_in[idx + 0];
            Bi[l][0] = (const float*)d_in[idx + 1];
            W[l][1] = (const float*)d_in[idx + 2];
            Bi[l][1] = (const float*)d_in[idx + 3];
            W[l][2] = (const float*)d_in[idx + 4];
            Bi[l][2] = (const float*)d_in[idx + 5];
            idx += 6;
        }
        dustbin = (const float*)d_in[35];
    }

    // Workspace layout (floats)
    float* ws    = (float*)d_ws;
    float* X0    = ws;                         // 1024*64
    float* Qb    = X0    + 1024 * 64;          // 1024*128
    float* Kb    = Qb    + 1024 * 128;
    float* Vb    = Kb    + 1024 * 128;
    float* HA    = Vb    + 1024 * 128;
    float* HB    = HA    + 1024 * 128;
    float* H5    = HB    + 1024 * 128;
    float* costs = H5    + 1024 * 128;         // 512*512
    float* Mm    = costs + 512 * 512;          // 513*516
    float* fv    = Mm    + N2 * MS;            // 513
    float* gv    = fv    + N2;                 // 513

    // 1) positional encoding + descriptor add -> X0 (1024x64)
    posenc_kernel<<<NNODE, 64, 0, stream>>>(p1, d1, p2, d2, fc1w, fc1b, fc2w, fc2b, X0);

    auto gemm_blocks = [](int M, int N) { return ((M / 16) * (N / 16) + 3) / 4; };

    // 2) four attentional layers (intra, cross, intra, cross)
    const float* layerIn[4]  = { X0, HA, HB, HA };
    float*       layerOut[4] = { HA, HB, HA, HB };
    const int    layerK[4]   = { 64, 128, 128, 128 };
    const int    layerCross[4] = { 0, 1, 0, 1 };
    for (int l = 0; l < 4; ++l) {
        int Kd = layerK[l];
        int nb = gemm_blocks(NNODE, CH);
        gemm_f32_wmma<0><<<nb, 128, 0, stream>>>(layerIn[l], Kd, W[l][0], CH, Bi[l][0],
                                                 Qb, CH, NNODE / 16, CH / 16, Kd, 0);
        gemm_f32_wmma<0><<<nb, 128, 0, stream>>>(layerIn[l], Kd, W[l][1], CH, Bi[l][1],
                                                 Kb, CH, NNODE / 16, CH / 16, Kd, 0);
        gemm_f32_wmma<0><<<nb, 128, 0, stream>>>(layerIn[l], Kd, W[l][2], CH, Bi[l][2],
                                                 Vb, CH, NNODE / 16, CH / 16, Kd, 0);
        const float* resid = (l == 0) ? nullptr : layerIn[l];
        if (layerCross[l])
            att_conv_kernel<1><<<NNODE / 8, 256, 0, stream>>>(Qb, Kb, Vb, resid, layerOut[l]);
        else
            att_conv_kernel<0><<<NNODE / 8, 256, 0, stream>>>(Qb, Kb, Vb, resid, layerOut[l]);
    }

    // 3) fc3 + relu -> H5, then L2 row-normalize
    gemm_f32_wmma<0><<<gemm_blocks(NNODE, CH), 128, 0, stream>>>(
        HB, CH, fc3w, CH, fc3b, H5, CH, NNODE / 16, CH / 16, CH, 1);
    row_normalize<<<NNODE, CH, 0, stream>>>(H5);

    // 4) costs = v1 @ v2^T  (512x128 @ 128x512, B = transposed view of H5[512:])
    gemm_f32_wmma<1><<<gemm_blocks(LEN, LEN), 128, 0, stream>>>(
        H5, CH, H5 + (size_t)LEN * CH, CH, nullptr, costs, LEN,
        LEN / 16, LEN / 16, CH, 0);

    // 5) M = 1 - cost_with_dustbin (513x513, stride MS)
    build_M<<<(N2 * N2 + 255) / 256, 256, 0, stream>>>(costs, dustbin, Mm);

    // 6) log-domain Sinkhorn, 100 iterations
    zero_fg<<<(N2 + 255) / 256, 256, 0, stream>>>(fv, gv);
    for (int it = 0; it < 100; ++it) {
        sinkhorn_row<<<N2, 128, 0, stream>>>(Mm, gv, fv);
        sinkhorn_col<<<N2, 128, 0, stream>>>(Mm, fv, gv);
    }

    // 7) gather matches, NLL -> d_out[0]
    loss_kernel<<<1, 512, 0, stream>>>(Mm, fv, gv, matches, (float*)d_out);
}